// LightGCN_8486855376918
// MI455X (gfx1250) — compile-verified
//
#include <hip/hip_runtime.h>
#include <hip/hip_bf16.h>

// ---------------------------------------------------------------------------
// LightGCN propagation for MI455X (gfx1250).
//
//   acc = e; x = e; repeat 3x: x = segment_sum(val * x[src] -> dst); acc += x
//   out = acc / 4
//
// Counting-sort the 4M edges by destination ONCE (integer atomics only), then
// each layer is a pure per-destination GATHER with register accumulation ->
// zero float atomics in the hot path.  The node-feature buffers (x, x_next,
// acc==d_out) total 192MB == the MI455X global L2, so gathers stay
// L2-resident; edge streams are loaded non-temporal so they can't evict them.
// ---------------------------------------------------------------------------

#define N_USERS   100000
#define N_ITEMS   150000
#define N_NODES   (N_USERS + N_ITEMS)          // 250000
#define N_EDGES   4000000
#define DIM       64
#define NODE_FLOATS ((size_t)N_NODES * DIM)    // 16,000,000 floats = 64 MB
#define USER_FLOATS ((size_t)N_USERS * DIM)    // 6,400,000 floats

#define THREADS     256
#define SCAN_BLOCKS ((N_NODES + THREADS - 1) / THREADS)   // 977 (<= 1024)

// ---------------------------------------------------------------------------
// k_init: x = concat(user_emb, item_emb); acc(out) = x
// ---------------------------------------------------------------------------
__global__ void k_init(const float* __restrict__ uemb,
                       const float* __restrict__ iemb,
                       float* __restrict__ x,
                       float* __restrict__ out) {
    size_t i = ((size_t)blockIdx.x * blockDim.x + threadIdx.x) * 4;
    if (i >= NODE_FLOATS) return;
    float4 v;
    if (i < USER_FLOATS) v = *(const float4*)(uemb + i);
    else                 v = *(const float4*)(iemb + (i - USER_FLOATS));
    *(float4*)(x + i)   = v;
    *(float4*)(out + i) = v;
}

// ---------------------------------------------------------------------------
// Counting sort by destination, step 1: zero bucket counters.
// ---------------------------------------------------------------------------
__global__ void k_zero_counts(int* __restrict__ counts) {
    int i = blockIdx.x * blockDim.x + threadIdx.x;
    if (i < N_NODES) counts[i] = 0;
}

// Step 2: histogram of edge_dst (integer atomics, 1 per edge).
__global__ void k_hist(const int* __restrict__ edst, int* __restrict__ counts) {
    int e = blockIdx.x * blockDim.x + threadIdx.x;
    if (e >= N_EDGES) return;
    int d = __builtin_nontemporal_load(edst + e);   // streaming, keep L2 clean
    atomicAdd(counts + d, 1);                       // global_atomic_add_u32
}

// Step 3a: per-block exclusive scan (LDS Hillis-Steele), emit block sums.
__global__ void k_scan1(const int* __restrict__ counts,
                        int* __restrict__ offsets,
                        int* __restrict__ bsum) {
    __shared__ int s[THREADS];
    const int tid = threadIdx.x;
    const int i   = blockIdx.x * THREADS + tid;
    int v = (i < N_NODES) ? counts[i] : 0;
    s[tid] = v;
    __syncthreads();
    for (int off = 1; off < THREADS; off <<= 1) {
        int t = (tid >= off) ? s[tid - off] : 0;
        __syncthreads();
        s[tid] += t;
        __syncthreads();
    }
    if (i < N_NODES) offsets[i] = s[tid] - v;       // exclusive within block
    if (tid == THREADS - 1) bsum[blockIdx.x] = s[tid];
}

// Step 3b: single-block exclusive scan of the 977 block sums.
__global__ void k_scan2(const int* __restrict__ bsum, int* __restrict__ bsumx) {
    __shared__ int s[1024];
    const int tid = threadIdx.x;
    int v = (tid < SCAN_BLOCKS) ? bsum[tid] : 0;
    s[tid] = v;
    __syncthreads();
    for (int off = 1; off < 1024; off <<= 1) {
        int t = (tid >= off) ? s[tid - off] : 0;
        __syncthreads();
        s[tid] += t;
        __syncthreads();
    }
    if (tid < SCAN_BLOCKS) bsumx[tid] = s[tid] - v; // exclusive
}

// Step 3c: add block offsets; init per-node cursors; terminate offsets[].
__global__ void k_scan3(int* __restrict__ offsets,
                        const int* __restrict__ bsumx,
                        int* __restrict__ cursors) {
    int i = blockIdx.x * blockDim.x + threadIdx.x;
    if (i < N_NODES) {
        int o = offsets[i] + bsumx[i >> 8];
        offsets[i] = o;
        cursors[i] = o;
    }
    if (i == 0) offsets[N_NODES] = N_EDGES;
}

// Step 4: place each edge into its destination bucket (cursor bump).
__global__ void k_place(const int*   __restrict__ esrc,
                        const int*   __restrict__ edst,
                        const float* __restrict__ evals,
                        int*   __restrict__ cursors,
                        int*   __restrict__ ssrc,
                        float* __restrict__ sval) {
    int e = blockIdx.x * blockDim.x + threadIdx.x;
    if (e >= N_EDGES) return;
    int   s = __builtin_nontemporal_load(esrc + e);
    int   d = __builtin_nontemporal_load(edst + e);
    float v = __builtin_nontemporal_load(evals + e);
    int pos = atomicAdd(cursors + d, 1);            // global_atomic_add_u32 RTN
    ssrc[pos] = s;
    sval[pos] = v;
}

// ---------------------------------------------------------------------------
// k_gather: one wave32 per destination node; lane l accumulates dims
// [2l, 2l+1] in registers.  Bucket bounds + sorted edge records are wave-
// uniform (readfirstlane) -> scalar s_loads, one fetch per wave per edge.
// Loop unrolled x2 with independent accumulators so two global_load_b64
// gathers are in flight before either FMA consumes them (MLP).
// Fuses the LightGCN accumulation: out = (out + rowsum) * scale.
// Final layer skips the x_next store (dead 64MB of bandwidth).
// NO float atomics anywhere.
// ---------------------------------------------------------------------------
__global__ void k_gather(const int*   __restrict__ offs,
                         const int*   __restrict__ ssrc,
                         const float* __restrict__ sval,
                         const float* __restrict__ x,
                         float*       __restrict__ xnext,
                         float*       __restrict__ out,
                         float scale,
                         int write_next) {
    const int lane = threadIdx.x & 31;
    const int node = __builtin_amdgcn_readfirstlane(
        (int)(blockIdx.x * (blockDim.x >> 5) + (threadIdx.x >> 5)));
    if (node >= N_NODES) return;

    const int beg = offs[node];
    const int end = offs[node + 1];

    // Warm the bucket's sorted-edge cachelines (global_prefetch_b8).
    if (beg < end) {
        __builtin_prefetch(ssrc + beg, 0, 3);
        __builtin_prefetch(sval + beg, 0, 3);
    }

    const size_t lo = (size_t)(lane * 2);
    float2 acc0 = make_float2(0.f, 0.f);
    float2 acc1 = make_float2(0.f, 0.f);

    int k = beg;
    for (; k + 1 < end; k += 2) {
        const int   s0 = ssrc[k];       // uniform -> scalar load
        const int   s1 = ssrc[k + 1];
        const float v0 = sval[k];
        const float v1 = sval[k + 1];
        const float2 m0 = *(const float2*)(x + (size_t)s0 * DIM + lo);
        const float2 m1 = *(const float2*)(x + (size_t)s1 * DIM + lo);
        acc0.x += v0 * m0.x;  acc0.y += v0 * m0.y;
        acc1.x += v1 * m1.x;  acc1.y += v1 * m1.y;
    }
    if (k < end) {
        const int   s0 = ssrc[k];
        const float v0 = sval[k];
        const float2 m0 = *(const float2*)(x + (size_t)s0 * DIM + lo);
        acc0.x += v0 * m0.x;  acc0.y += v0 * m0.y;
    }
    acc0.x += acc1.x;
    acc0.y += acc1.y;

    const size_t ro = (size_t)node * DIM + lo;
    if (write_next) {
        *(float2*)(xnext + ro) = acc0;              // x for next layer
    }
    float2 o = *(const float2*)(out + ro);
    o.x = (o.x + acc0.x) * scale;
    o.y = (o.y + acc0.y) * scale;
    *(float2*)(out + ro) = o;
}

// ---------------------------------------------------------------------------
// kernel_launch
// inputs (setup_inputs order): user_emb, item_emb, edge_vals, edge_src, edge_dst
// d_out: 250000*64 floats = concat(users, items) = full light_out matrix.
// ws layout: xA(64MB) | xB(64MB) | sval(16MB) | ssrc(16MB) | counts | offsets
//            | cursors | bsum | bsumx   (~163 MB total)
// ---------------------------------------------------------------------------
extern "C" void kernel_launch(void* const* d_in, const int* in_sizes, int n_in,
                              void* d_out, int out_size, void* d_ws, size_t ws_size,
                              hipStream_t stream) {
    const float* uemb  = (const float*)d_in[0];
    const float* iemb  = (const float*)d_in[1];
    const float* evals = (const float*)d_in[2];
    const int*   esrc  = (const int*)d_in[3];
    const int*   edst  = (const int*)d_in[4];
    float* out = (float*)d_out;

    float* xA      = (float*)d_ws;
    float* xB      = xA + NODE_FLOATS;
    float* sval    = xB + NODE_FLOATS;
    int*   ssrc    = (int*)(sval + N_EDGES);
    int*   counts  = ssrc + N_EDGES;
    int*   offsets = counts + N_NODES;          // N_NODES + 1 entries
    int*   cursors = offsets + (N_NODES + 1);
    int*   bsum    = cursors + N_NODES;
    int*   bsumx   = bsum + 1024;

    const int elem_blocks = (int)((NODE_FLOATS / 4 + THREADS - 1) / THREADS); // 15625
    const int edge_blocks = (N_EDGES + THREADS - 1) / THREADS;                // 15625
    const int node_blocks = SCAN_BLOCKS;                                      // 977
    const int wave_blocks = (N_NODES + (THREADS / 32) - 1) / (THREADS / 32);  // 31250

    // Embedding init (independent of the sort chain).
    k_init<<<elem_blocks, THREADS, 0, stream>>>(uemb, iemb, xA, out);

    // Counting sort of edges by destination (integer atomics only).
    k_zero_counts<<<node_blocks, THREADS, 0, stream>>>(counts);
    k_hist<<<edge_blocks, THREADS, 0, stream>>>(edst, counts);
    k_scan1<<<node_blocks, THREADS, 0, stream>>>(counts, offsets, bsum);
    k_scan2<<<1, 1024, 0, stream>>>(bsum, bsumx);
    k_scan3<<<node_blocks, THREADS, 0, stream>>>(offsets, bsumx, cursors);
    k_place<<<edge_blocks, THREADS, 0, stream>>>(esrc, edst, evals, cursors, ssrc, sval);

    // Three propagation layers: pure gather, accumulation fused.
    float* cur = xA;
    float* nxt = xB;
    for (int layer = 0; layer < 3; ++layer) {
        const float scale      = (layer == 2) ? 0.25f : 1.0f;
        const int   write_next = (layer == 2) ? 0 : 1;
        k_gather<<<wave_blocks, THREADS, 0, stream>>>(offsets, ssrc, sval,
                                                      cur, nxt, out,
                                                      scale, write_next);
        float* t = cur; cur = nxt; nxt = t;
    }
}